// BasicBlock_32830730010863
// MI455X (gfx1250) — compile-verified
//
#include <hip/hip_runtime.h>
#include <math.h>

#ifndef __has_builtin
#define __has_builtin(x) 0
#endif

// ---------------- problem geometry ----------------
#define NB 4
#define NC 3
#define DH 512
#define DW 512
#define HW (DH * DW)          // 262144
#define FIELD ((size_t)NB * NC * HW)   // 3145728
#define DARK  ((size_t)NB * HW)        // 1048576
#define RPAD 17
#define PATCH 35
#define CHUNKS 32
#define CHUNK_LEN (HW / CHUNKS)        // 8192 floats
#define CHUNK_V4 (CHUNK_LEN / 4)       // 2048 float4

// output layout (floats), tuple order:
// Bn, tn, Jn, G, Hn, Pn, Qn, un, vn, Xn, Yn, gamma_3
#define OFF_BN ((size_t)0)
#define OFF_TN (OFF_BN + FIELD)
#define OFF_JN (OFF_TN + FIELD)
#define OFF_G  (OFF_JN + FIELD)
#define OFF_HN (OFF_G  + FIELD)
#define OFF_P  (OFF_HN + FIELD)
#define OFF_Q  (OFF_P  + DARK)
#define OFF_UN (OFF_Q  + DARK)
#define OFF_VN (OFF_UN + FIELD)
#define OFF_XN (OFF_VN + FIELD)
#define OFF_YN (OFF_XN + FIELD)
#define OFF_G3 (OFF_YN + FIELD)

#define AS1 __attribute__((address_space(1)))
#define AS3 __attribute__((address_space(3)))

#if __has_builtin(__builtin_amdgcn_global_load_async_to_lds_b32)
#define HAVE_ASYNC_LDS 1
#else
#define HAVE_ASYNC_LDS 0
#endif

static __device__ __forceinline__ void wait_async0() {
#if __has_builtin(__builtin_amdgcn_s_wait_asynccnt)
  __builtin_amdgcn_s_wait_asynccnt(0);
#elif HAVE_ASYNC_LDS
  asm volatile("s_wait_asynccnt 0" ::: "memory");
#endif
}

static __device__ __forceinline__ int reflect_idx(int p, int n) {
  // jnp.pad(..., mode='reflect'): -1 -> 1, n -> n-2
  return p < 0 ? -p : (p >= n ? 2 * n - 2 - p : p);
}

static __device__ __forceinline__ float block_reduce_sum(float s) {
  __shared__ float red[256];
  red[threadIdx.x] = s;
  __syncthreads();
  for (int off = 128; off > 0; off >>= 1) {
    if ((int)threadIdx.x < off) red[threadIdx.x] += red[threadIdx.x + off];
    __syncthreads();
  }
  return red[0];
}

// ---------------- K1: partial sums of J per (b,c) chunk ----------------
// grid = 12*CHUNKS blocks; float4 loads, 2 loads in flight per iteration.
__global__ void k_meanJ(const float* __restrict__ J, float* __restrict__ partialJ) {
  int blk = blockIdx.x;
  int bc = blk / CHUNKS, ch = blk % CHUNKS;
  const float4* p =
      (const float4*)(J + (size_t)bc * HW + (size_t)ch * CHUNK_LEN);
  float s0 = 0.f, s1 = 0.f;
  for (int i = threadIdx.x; i < CHUNK_V4; i += 512) {
    float4 a = p[i];
    float4 b = p[i + 256];
    s0 += (a.x + a.y) + (a.z + a.w);
    s1 += (b.x + b.y) + (b.z + b.w);
  }
  float tot = block_reduce_sum(s0 + s1);
  if (threadIdx.x == 0) partialJ[bc * CHUNKS + ch] = tot;
}

// ---------------- K2: finalize means, per-batch argsort -> remap coeffs ----------------
__global__ void k_sort(const float* __restrict__ partialJ, float* __restrict__ coeff,
                       int* __restrict__ lidx, const float* __restrict__ g3,
                       float* __restrict__ out_g3) {
  int b = threadIdx.x;
  if (b < NB) {
    float m[3];
#pragma unroll
    for (int c = 0; c < 3; ++c) {
      float s = 0.f;
      for (int ch = 0; ch < CHUNKS; ++ch) s += partialJ[(b * 3 + c) * CHUNKS + ch];
      m[c] = s * (1.0f / (float)HW);
    }
    float v0 = m[0], v1 = m[1], v2 = m[2];
    int i0 = 0, i1 = 1, i2 = 2;
    float tv; int ti;
    if (v0 > v1) { tv = v0; v0 = v1; v1 = tv; ti = i0; i0 = i1; i1 = ti; }
    if (v1 > v2) { tv = v1; v1 = v2; v2 = tv; ti = i1; i1 = i2; i2 = ti; }
    if (v0 > v1) { tv = v0; v0 = v1; v1 = tv; ti = i0; i0 = i1; i1 = ti; }
    // ascending: (v0,i0) smallest ... (v2,i2) largest
    coeff[b * 3 + i2] = 0.0f;       // largest-mean channel unchanged
    coeff[b * 3 + i1] = v2 - v1;    // J_m = J[i1] + (ms2-ms1)*J_l
    coeff[b * 3 + i0] = v2 - v0;    // J_s = J[i0] + (ms2-ms0)*J_l
    lidx[b] = i2;
  }
  if (threadIdx.x == 0) out_g3[0] = g3[0];
}

// ---------------- K3: partial sums of B update per (b,c) chunk ----------------
__global__ void k_Bmean(const float* __restrict__ I, const float* __restrict__ Bp,
                        const float* __restrict__ t, const float* __restrict__ J,
                        const float* __restrict__ coeff, const int* __restrict__ lidx,
                        float* __restrict__ partialB) {
  int blk = blockIdx.x;
  int bc = blk / CHUNKS, ch = blk % CHUNKS;
  int b = bc / 3;
  float cf = coeff[bc];
  int li = lidx[b];
  size_t off = (size_t)bc * HW + (size_t)ch * CHUNK_LEN;
  size_t offl = (size_t)(b * 3 + li) * HW + (size_t)ch * CHUNK_LEN;
  const float4* Jc  = (const float4*)(J  + off);
  const float4* Jl  = (const float4*)(J  + offl);
  const float4* Ic  = (const float4*)(I  + off);
  const float4* Bpc = (const float4*)(Bp + off);
  const float4* tc  = (const float4*)(t  + off);
  float s = 0.f;
#pragma unroll 2
  for (int i = threadIdx.x; i < CHUNK_V4; i += 256) {
    float4 j4 = Jc[i], jl4 = Jl[i], i4 = Ic[i], bp4 = Bpc[i], t4 = tc[i];
    float jv[4] = {j4.x, j4.y, j4.z, j4.w};
    float lv[4] = {jl4.x, jl4.y, jl4.z, jl4.w};
    float iv[4] = {i4.x, i4.y, i4.z, i4.w};
    float bv[4] = {bp4.x, bp4.y, bp4.z, bp4.w};
    float tvv[4] = {t4.x, t4.y, t4.z, t4.w};
#pragma unroll
    for (int k = 0; k < 4; ++k) {
      float Jp  = fmaf(cf, lv[k], jv[k]);
      float omt = 1.0f - tvv[k];
      float num = 0.3f * bv[k] - (Jp * tvv[k] - iv[k]) * omt;  // lam1=1, lam3=0.3
      float den = fmaf(omt, omt, 0.3f);
      s += num / den;
    }
  }
  float tot = block_reduce_sum(s);
  if (threadIdx.x == 0) partialB[bc * CHUNKS + ch] = tot;
}

// ---------------- K3b: finalize Bmean ----------------
__global__ void k_finalB(const float* __restrict__ partialB, float* __restrict__ Bmean) {
  int bc = threadIdx.x;
  if (bc < NB * NC) {
    float s = 0.f;
    for (int ch = 0; ch < CHUNKS; ++ch) s += partialB[bc * CHUNKS + ch];
    Bmean[bc] = s * (1.0f / (float)HW);
  }
}

// ---------------- K4: fused elementwise (tn, Jn, un, vn, Yn, Bn, G, cmins) ----------------
__global__ void k_main(const float* __restrict__ I, const float* __restrict__ t_p,
                       const float* __restrict__ Hin, const float* __restrict__ Xin,
                       const float* __restrict__ J, const float* __restrict__ G,
                       const float* __restrict__ u, const float* __restrict__ v,
                       const float* __restrict__ Y,
                       const float* __restrict__ coeff, const int* __restrict__ lidx,
                       const float* __restrict__ Bmean, const float* __restrict__ t1w,
                       const float* __restrict__ g1p, const float* __restrict__ g2p,
                       const float* __restrict__ g3p, const float* __restrict__ g4p,
                       const float* __restrict__ g5p, float* __restrict__ out) {
  int pix = blockIdx.x * 256 + threadIdx.x;
  if (pix >= NB * HW) return;
  int b = pix >> 18;           // / HW
  int p = pix & (HW - 1);
  float g1 = g1p[0], g2 = g2p[0], g3 = g3p[0], g4 = g4p[0], g5 = g5p[0];
  int li = lidx[b];
  size_t base = (size_t)b * 3 * HW + p;
  float Jl = J[(size_t)(b * 3 + li) * HW + p];

  float tr[3], Iv[3], Bm[3], Gv[3], uv[3], vv[3], Yv[3];
#pragma unroll
  for (int c = 0; c < 3; ++c) {
    size_t idx = base + (size_t)c * HW;
    float Jc = J[idx] + coeff[b * 3 + c] * Jl;   // remapped J'
    Iv[c] = I[idx];
    Bm[c] = Bmean[b * 3 + c];
    float d = Jc - Bm[c];
    // tn numerator/denominator (lam1=1, lam2=0.7)
    tr[c] = (0.7f * t_p[idx] + g4 * Hin[idx] - (Bm[c] - Iv[c]) * d - Xin[idx]) /
            (d * d + 0.7f + g4);
    Gv[c] = G[idx]; uv[c] = u[idx]; vv[c] = v[idx]; Yv[c] = Y[idx];
  }
  float t1 = t1w[0] * tr[0] + t1w[1] * tr[1] + t1w[2] * tr[2];  // 1x1 conv, replicated
  float Jden = t1 * t1 + g3 + g4 + g5;
  float ru = 1.0f / (g1 + g4), rv = 1.0f / (g2 + g5);
  float cminu = INFINITY, cminv = INFINITY;
#pragma unroll
  for (int c = 0; c < 3; ++c) {
    size_t idx = base + (size_t)c * HW;
    float Jn = (t1 * (Iv[c] - Bm[c] * (1.0f - t1)) + g3 * Gv[c] + g4 * uv[c] -
                g5 * vv[c] - Yv[c] + g5) / Jden;
    float un = (g1 * uv[c] + g4 * Jn) * ru;
    float vn = (g2 * vv[c] - g5 * Jn + g5) * rv;
    out[OFF_TN + idx] = t1;                                   // reread by conv: keep RT
    __builtin_nontemporal_store(Bm[c], &out[OFF_BN + idx]);   // write-once streams: NT
    __builtin_nontemporal_store(Jn,    &out[OFF_JN + idx]);
    __builtin_nontemporal_store(Gv[c], &out[OFF_G  + idx]);
    __builtin_nontemporal_store(un,    &out[OFF_UN + idx]);
    __builtin_nontemporal_store(vn,    &out[OFF_VN + idx]);
    __builtin_nontemporal_store(Yv[c] + g3 * (Jn - Gv[c]), &out[OFF_YN + idx]);
    cminu = fminf(cminu, un);
    cminv = fminf(cminv, vn);
  }
  // channel-min staged into P/Q slots (reread by hmin: regular stores)
  out[OFF_P + (size_t)b * HW + p] = cminu;
  out[OFF_Q + (size_t)b * HW + p] = cminv;
}

// ---------------- K5: 3x3 conv (zero pad) -> Hn, Xn ----------------
__global__ void k_conv(const float* __restrict__ Xin, const float* __restrict__ wco,
                       const float* __restrict__ g4p, float* __restrict__ out) {
  int pix = blockIdx.x * 256 + threadIdx.x;
  if (pix >= NB * HW) return;
  int b = pix >> 18, p = pix & (HW - 1);
  int y = p >> 9, x = p & (DW - 1);
  float g4 = g4p[0], rg4 = 1.0f / g4;
  const float* tn = out + OFF_TN + (size_t)b * 3 * HW;  // channel 0 (all equal)
  const float* Xb = Xin + (size_t)b * 3 * HW;
  float acc0 = 0.f, acc1 = 0.f, acc2 = 0.f;
#pragma unroll
  for (int kh = 0; kh < 3; ++kh) {
    int yy = y + kh - 1;
    if ((unsigned)yy >= DH) continue;
#pragma unroll
    for (int kw = 0; kw < 3; ++kw) {
      int xx = x + kw - 1;
      if ((unsigned)xx >= DW) continue;
      int np = (yy << 9) | xx;
      float t1 = tn[np];
#pragma unroll
      for (int c = 0; c < 3; ++c) {
        float a = fmaf(Xb[(size_t)c * HW + np], rg4, t1);  // tn + X/g4, zero outside
        acc0 = fmaf(wco[((0 * 3 + c) * 3 + kh) * 3 + kw], a, acc0);
        acc1 = fmaf(wco[((1 * 3 + c) * 3 + kh) * 3 + kw], a, acc1);
        acc2 = fmaf(wco[((2 * 3 + c) * 3 + kh) * 3 + kw], a, acc2);
      }
    }
  }
  float t1c = tn[p];
  float acc[3] = {acc0, acc1, acc2};
#pragma unroll
  for (int o = 0; o < 3; ++o) {
    size_t idx = (size_t)(b * 3 + o) * HW + p;
    __builtin_nontemporal_store(acc[o], &out[OFF_HN + idx]);
    __builtin_nontemporal_store(Xin[idx] + g4 * (t1c - acc[o]), &out[OFF_XN + idx]);
  }
}

// ---------------- K6: horizontal 35-window min (reflect), async global->LDS ----------------
__global__ void k_hmin(const float* __restrict__ inU, const float* __restrict__ inV,
                       float* __restrict__ tmpU, float* __restrict__ tmpV) {
  int row = blockIdx.x;  // 0 .. NB*DH-1
  const float* in = blockIdx.y ? inV : inU;
  float* outp = blockIdx.y ? tmpV : tmpU;
  __shared__ float lds[DW + 2 * RPAD];  // 546 floats
  const float* rowp = in + (size_t)row * DW;
  for (int i = threadIdx.x; i < DW + 2 * RPAD; i += 256) {
    int sx = reflect_idx(i - RPAD, DW);
#if HAVE_ASYNC_LDS
    __builtin_amdgcn_global_load_async_to_lds_b32(
        (AS1 int*)(rowp + sx), (AS3 int*)&lds[i], 0, 0);
#else
    lds[i] = rowp[sx];
#endif
  }
  wait_async0();
  __syncthreads();
  for (int x = threadIdx.x; x < DW; x += 256) {
    float m = lds[x];
#pragma unroll
    for (int k = 1; k < PATCH; ++k) m = fminf(m, lds[x + k]);
    outp[(size_t)row * DW + x] = m;
  }
}

// ---------------- K7: vertical 35-window min (reflect) + soft threshold ----------------
__global__ void k_vmin(const float* __restrict__ tmpU, const float* __restrict__ tmpV,
                       const float* __restrict__ g1p, float* __restrict__ out) {
  int pix = blockIdx.x * 256 + threadIdx.x;
  if (pix >= NB * HW) return;
  int b = pix >> 18, p = pix & (HW - 1);
  int y = p >> 9, x = p & (DW - 1);
  float mu = INFINITY, mv = INFINITY;
#pragma unroll 7
  for (int k = 0; k < PATCH; ++k) {
    int yy = reflect_idx(y + k - RPAD, DH);
    size_t idx = ((size_t)b * DH + yy) * DW + x;   // coalesced across x
    mu = fminf(mu, tmpU[idx]);
    mv = fminf(mv, tmpV[idx]);
  }
  float lam = 1.0f / g1p[0];  // lam4 = lam5 = 1.0
  float au = fmaxf(fabsf(mu) - lam, 0.0f);
  float av = fmaxf(fabsf(mv) - lam, 0.0f);
  __builtin_nontemporal_store(copysignf(au, mu), &out[OFF_P + (size_t)b * HW + p]);
  __builtin_nontemporal_store(copysignf(av, mv), &out[OFF_Q + (size_t)b * HW + p]);
}

// ---------------- launcher ----------------
extern "C" void kernel_launch(void* const* d_in, const int* in_sizes, int n_in,
                              void* d_out, int out_size, void* d_ws, size_t ws_size,
                              hipStream_t stream) {
  const float* I   = (const float*)d_in[0];
  const float* t_p = (const float*)d_in[1];
  const float* B_p = (const float*)d_in[2];
  // d_in[3] = B (unused by reference)
  const float* t   = (const float*)d_in[4];
  const float* J   = (const float*)d_in[5];
  const float* G   = (const float*)d_in[6];
  const float* Hh  = (const float*)d_in[7];
  // d_in[8] = P, d_in[9] = Q (unused)
  const float* u   = (const float*)d_in[10];
  const float* v   = (const float*)d_in[11];
  const float* X   = (const float*)d_in[12];
  const float* Y   = (const float*)d_in[13];
  const float* g1  = (const float*)d_in[14];
  const float* g2  = (const float*)d_in[15];
  const float* g3  = (const float*)d_in[16];
  const float* g4  = (const float*)d_in[17];
  const float* g5  = (const float*)d_in[18];
  const float* t1w = (const float*)d_in[19];
  const float* hwc = (const float*)d_in[20];
  // d_in[21] = patch_size (35, compile-time here)

  char* ws = (char*)d_ws;
  float* coeff    = (float*)(ws + 0);
  int*   lidx     = (int*)(ws + 64);
  float* Bmean    = (float*)(ws + 128);
  float* partialJ = (float*)(ws + 256);    // 12*32 floats
  float* partialB = (float*)(ws + 2048);   // 12*32 floats
  float* tmpU     = (float*)(ws + 4096);
  float* tmpV     = (float*)(ws + 4096 + sizeof(float) * DARK);

  float* out = (float*)d_out;
  const int nPix = NB * HW;

  k_meanJ<<<NB * NC * CHUNKS, 256, 0, stream>>>(J, partialJ);
  k_sort<<<1, 32, 0, stream>>>(partialJ, coeff, lidx, g3, out + OFF_G3);
  k_Bmean<<<NB * NC * CHUNKS, 256, 0, stream>>>(I, B_p, t, J, coeff, lidx, partialB);
  k_finalB<<<1, 32, 0, stream>>>(partialB, Bmean);
  k_main<<<nPix / 256, 256, 0, stream>>>(I, t_p, Hh, X, J, G, u, v, Y,
                                         coeff, lidx, Bmean, t1w,
                                         g1, g2, g3, g4, g5, out);
  k_conv<<<nPix / 256, 256, 0, stream>>>(X, hwc, g4, out);
  k_hmin<<<dim3(NB * DH, 2), 256, 0, stream>>>(out + OFF_P, out + OFF_Q, tmpU, tmpV);
  k_vmin<<<nPix / 256, 256, 0, stream>>>(tmpU, tmpV, g1, out);
}